// MyConv2D_37692632989867
// MI455X (gfx1250) — compile-verified
//
#include <hip/hip_runtime.h>

// CDNA5 / gfx1250 implicit-GEMM 3x3 conv (fp32) via V_WMMA_F32_16X16X4_F32.
//
// GEMM view: [M = N*H*W pixels] x [K = Cin*3*3 = 288] x [N = Cout = 32].
// Register blocking: each wave32 computes a 64(M) x 32(N) tile as EIGHT 16x16
// WMMA accumulators (4 M-tiles x 2 N-tiles). Per K=4 step: 8 WMMAs fed by
// 4 A-loads + 2 B-loads (0.75 ds_load_b64 per WMMA; B reused across M-tiles,
// 8 independent accumulate chains hide WMMA RAW latency).
//
// Block tile: 4 output rows x 128 cols (8 waves = 4 rows x 2 x-segments).
// Input stage: 6 rows x 130 cols x 32ch -> 1.5x HBM read amplification;
// kernel is fp32-WMMA compute-bound (~125us compute vs ~29us HBM), so LDS
// occupancy of 2 blocks/WGP (16 waves) is sufficient.

typedef __attribute__((ext_vector_type(2))) float v2f;
typedef __attribute__((ext_vector_type(8))) float v8f;

#define BLOCK_THREADS 256          // 8 waves (wave32)
#define ROWS          4            // output rows per block
#define COLS          128          // output cols per block
#define X_DIM         130          // halo cols: x0-1 .. x0+128
#define CI_STR        34           // ci stride pad 32->34: conflict-free A reads
#define LDS_IN        (6 * X_DIM * CI_STR)   // 26520 floats (~106 KB)
#define LDS_WT        (9 * 32 * 32)          // 9216 floats  (36 KB)

__global__ __launch_bounds__(BLOCK_THREADS)
void conv3x3_wmma_f32(const float* __restrict__ xg,   // (8,32,512,512) NCHW
                      const float* __restrict__ wg,   // (32,32,3,3)    OIHW
                      const float* __restrict__ bg,   // (32,)
                      float* __restrict__ og)         // (8,32,512,512) NCHW
{
    __shared__ float lin[LDS_IN];  // input: [row 0..5][x 0..129][ci], ci-stride 34
    __shared__ float lwt[LDS_WT];  // weights: [tap][ci>>1][o][ci&1] (K-pairs adjacent)

    const int tid  = threadIdx.x;
    const int bid  = blockIdx.x;            // 8 * 128 * 4 = 4096 blocks
    const int cblk = bid & 3;
    const int rblk = (bid >> 2) & 127;
    const int nb   = bid >> 9;
    const int x0   = cblk * COLS;
    const int y0   = rblk * ROWS;

    // ---- Stage weights: coalesced global read (t == OIHW linear index),
    //      scatter into [tap][ci-pair][o][par] so a B fragment is one b64 load.
    for (int t = tid; t < LDS_WT; t += BLOCK_THREADS) {
        const int tap = t % 9;              // kh*3+kw
        const int ci  = (t / 9) & 31;
        const int o   = t / 288;
        lwt[(((tap * 16 + (ci >> 1)) * 32 + o) << 1) + (ci & 1)] = wg[t];
    }

    // ---- Stage input tile (rows y0-1..y0+4, cols x0-1..x0+128, 32 ch),
    //      zero-filling padding. Global reads coalesced along x.
    for (int t = tid; t < 6 * 32 * X_DIM; t += BLOCK_THREADS) {
        const int xx = t % X_DIM;
        const int ci = (t / X_DIM) & 31;
        const int r  = t / (32 * X_DIM);    // 0..5
        const int gh = y0 - 1 + r;
        const int gw = x0 - 1 + xx;
        float v = 0.0f;
        if ((unsigned)gh < 512u && (unsigned)gw < 512u)
            v = xg[(((nb * 32 + ci) * 512 + gh) << 9) + gw];
        lin[(r * X_DIM + xx) * CI_STR + ci] = v;
    }
    __syncthreads();

    const int tid5 = threadIdx.x;
    const int lane = tid5 & 31;
    const int wave = tid5 >> 5;
    const int row  = wave >> 1;             // 0..3: output row within block
    const int xseg = wave & 1;              // 0..1: 64-pixel x segment
    const int m    = lane & 15;
    const int hi   = lane >> 4;             // half-wave selects K-pair (ISA A layout)

    v8f acc00 = {}, acc01 = {};             // M-tile 0, ch 0..15 / 16..31
    v8f acc10 = {}, acc11 = {};             // M-tile 1
    v8f acc20 = {}, acc21 = {};             // M-tile 2
    v8f acc30 = {}, acc31 = {};             // M-tile 3

    #pragma unroll
    for (int kh = 0; kh < 3; ++kh) {
        #pragma unroll
        for (int kw = 0; kw < 3; ++kw) {
            const int tap = kh * 3 + kw;
            const float* arow0 =
                &lin[((row + kh) * X_DIM + xseg * 64 + m + kw) * CI_STR];
            const float* arow1 = arow0 + 16 * CI_STR;
            const float* arow2 = arow0 + 32 * CI_STR;
            const float* arow3 = arow0 + 48 * CI_STR;
            const float* brow  = &lwt[tap * 1024];
            #pragma unroll
            for (int c0 = 0; c0 < 32; c0 += 4) {
                const int kb = c0 + (hi << 1);
                v2f a0 = *(const v2f*)&arow0[kb];                              // ds b64
                v2f a1 = *(const v2f*)&arow1[kb];
                v2f a2 = *(const v2f*)&arow2[kb];
                v2f a3 = *(const v2f*)&arow3[kb];
                v2f b0 = *(const v2f*)&brow[((c0 >> 1) + hi) * 64 + (m << 1)];
                v2f b1 = *(const v2f*)&brow[((c0 >> 1) + hi) * 64 + ((m + 16) << 1)];
                acc00 = __builtin_amdgcn_wmma_f32_16x16x4_f32(
                            false, a0, false, b0, (short)0, acc00, false, false);
                acc01 = __builtin_amdgcn_wmma_f32_16x16x4_f32(
                            false, a0, false, b1, (short)0, acc01, false, false);
                acc10 = __builtin_amdgcn_wmma_f32_16x16x4_f32(
                            false, a1, false, b0, (short)0, acc10, false, false);
                acc11 = __builtin_amdgcn_wmma_f32_16x16x4_f32(
                            false, a1, false, b1, (short)0, acc11, false, false);
                acc20 = __builtin_amdgcn_wmma_f32_16x16x4_f32(
                            false, a2, false, b0, (short)0, acc20, false, false);
                acc21 = __builtin_amdgcn_wmma_f32_16x16x4_f32(
                            false, a2, false, b1, (short)0, acc21, false, false);
                acc30 = __builtin_amdgcn_wmma_f32_16x16x4_f32(
                            false, a3, false, b0, (short)0, acc30, false, false);
                acc31 = __builtin_amdgcn_wmma_f32_16x16x4_f32(
                            false, a3, false, b1, (short)0, acc31, false, false);
            }
        }
    }

    // ---- Epilogue: bias + store. D layout: VGPR j -> (M = j + 8*hi, N = m).
    const float bias0 = bg[m];
    const float bias1 = bg[m + 16];
    const int   gy    = y0 + row;
    const int   wbase = x0 + xseg * 64 + (hi << 3);
    float* p00 = og + (((nb * 32 + m)      * 512 + gy) << 9) + wbase;   // ch lo
    float* p01 = og + (((nb * 32 + m + 16) * 512 + gy) << 9) + wbase;   // ch hi
    #pragma unroll
    for (int j = 0; j < 8; ++j) {
        p00[j]      = acc00[j] + bias0;   // contiguous -> b128 stores
        p00[j + 16] = acc10[j] + bias0;
        p00[j + 32] = acc20[j] + bias0;
        p00[j + 48] = acc30[j] + bias0;
        p01[j]      = acc01[j] + bias1;
        p01[j + 16] = acc11[j] + bias1;
        p01[j + 32] = acc21[j] + bias1;
        p01[j + 48] = acc31[j] + bias1;
    }
}

extern "C" void kernel_launch(void* const* d_in, const int* in_sizes, int n_in,
                              void* d_out, int out_size, void* d_ws, size_t ws_size,
                              hipStream_t stream) {
    const float* x = (const float*)d_in[0];
    const float* w = (const float*)d_in[1];
    const float* b = (const float*)d_in[2];
    float* out = (float*)d_out;
    (void)in_sizes; (void)n_in; (void)out_size; (void)d_ws; (void)ws_size;

    // grid: 8 batches * 128 row-blocks * 4 col-blocks = 4096 blocks, 256 thr
    dim3 grid(8 * 128 * 4);
    conv3x3_wmma_f32<<<grid, BLOCK_THREADS, 0, stream>>>(x, w, b, out);
}